// CVCSolver_56839597195424
// MI455X (gfx1250) — compile-verified
//
#include <hip/hip_runtime.h>
#include <hip/hip_bf16.h>
#include <stdint.h>

#define PI_F 3.14159265358979323846f

typedef float v2f __attribute__((ext_vector_type(2)));
typedef float v8f __attribute__((ext_vector_type(8)));

// ---------------------------------------------------------------------------
// Kernel 1: fine-grid velocity interpolation as a WMMA GEMM.
//   Vf[64][512] = vxs[64][101] @ W[104][512]   (K padded 101->104, N 501->512)
// A-panel (16 rows x 101 K) is contiguous in vxs -> staged to LDS with
// global_load_async_to_lds_b128; K-loop is branchless ds_load + wmma.
// ---------------------------------------------------------------------------
__global__ __launch_bounds__(32) void cvc_interp_wmma(const float* __restrict__ vxs,
                                                      float* __restrict__ Vf) {
  // 16*101 = 1616 panel floats + 8 zero-pad floats (row 15's K=101..103 reads)
  __shared__ __align__(16) float sA[1624];

  const int tile = blockIdx.x;          // 128 tiles = 4 (M) x 32 (N)
  const int tm   = (tile >> 5) << 4;
  const int tn   = (tile & 31) << 4;
  const int lane = threadIdx.x;
  const int half = lane >> 4;           // 0: K kc+0/1, 1: K kc+2/3 (16x4 f32 layout)
  const int l    = lane & 15;

  // ---- async stage of the contiguous 6464-byte A-panel (404 b128 chunks) ----
  {
    const uint64_t gbase = (uint64_t)(uintptr_t)(vxs + tm * 101);
    const uint32_t lbase = (uint32_t)(uintptr_t)(&sA[0]);
    for (int idx = lane; idx < 404; idx += 32) {
      uint32_t lds = lbase + (uint32_t)idx * 16u;
      uint64_t ga  = gbase + (uint64_t)idx * 16u;
      asm volatile("global_load_async_to_lds_b128 %0, %1, off"
                   :: "v"(lds), "v"(ga) : "memory");
    }
    if (lane < 8) sA[1616 + lane] = 0.f;          // zero tail: avoid NaN*0
    asm volatile("s_wait_asynccnt 0" ::: "memory");
  }
  __syncthreads();

  // ---- per-lane B-column parameters (branchless weights) ----
  const int n  = tn + l;                           // fine-grid column
  int i0 = (n <= 500) ? ((n / 5 > 99) ? 99 : n / 5) : -2;  // -2: all-zero column
  const float w = (float)n * 0.2f - (float)i0;

  v8f c = {0.f, 0.f, 0.f, 0.f, 0.f, 0.f, 0.f, 0.f};
#pragma unroll
  for (int kc = 0; kc < 104; kc += 4) {
    const int ka = kc + half * 2;
    v2f a, b;
    a.x = sA[l * 101 + ka];                        // conflict-free ds reads
    a.y = sA[l * 101 + ka + 1];
    b.x = (ka     == i0) ? (1.f - w) : ((ka     == i0 + 1) ? w : 0.f);
    b.y = (ka + 1 == i0) ? (1.f - w) : ((ka + 1 == i0 + 1) ? w : 0.f);
    c = __builtin_amdgcn_wmma_f32_16x16x4_f32(false, a, false, b, (short)0, c,
                                              false, false);
  }
  // D layout: VGPR r -> M = r (+8 for lanes 16-31), N = lane&15
#pragma unroll
  for (int r = 0; r < 8; ++r)
    Vf[(size_t)(tm + half * 8 + r) * 512 + (tn + l)] = c[r];
}

// ---------------------------------------------------------------------------
// Kernel 2: latency-optimal solver. One wave per batch; all state in VGPRs.
// Dense 500x500 t_K matvec collapses to v'[i] = -k_i v'[i-1] + buf[i];
// parallelized as an affine (P,S) wave scan: 16 elems/lane + 5-level scan.
// ---------------------------------------------------------------------------
__device__ __forceinline__ float lane_up(float x, int d, float ident, int lane) {
  int v = __builtin_amdgcn_ds_bpermute((lane - d) << 2, __float_as_int(x));
  float r = __int_as_float(v);
  return (lane < d) ? ident : r;
}

__global__ __launch_bounds__(32) void cvc_solver(const float* __restrict__ Vf,
                                                 float* __restrict__ out) {
  __shared__ __align__(16) float sVf[512];
  __shared__ float sTb[500];
  const int b    = blockIdx.x;
  const int lane = threadIdx.x;
  const size_t ob = (size_t)b * 10201u;          // 101*101 per batch

  // ---- async stage fine-velocity row into LDS ----
  {
    const float* src = Vf + (size_t)b * 512;
#pragma unroll
    for (int r = 0; r < 4; ++r) {
      uint32_t lds = (uint32_t)(uintptr_t)(&sVf[r * 128 + lane * 4]);
      uint64_t ga  = (uint64_t)(uintptr_t)(src + r * 128 + lane * 4);
      asm volatile("global_load_async_to_lds_b128 %0, %1, off"
                   :: "v"(lds), "v"(ga) : "memory");
    }
    asm volatile("s_wait_asynccnt 0" ::: "memory");
  }
  __syncthreads();

  // ---- per-lane time-invariant state: k, cumulative products p, initial v ----
  float kk[16], pp[16], vv[16], ss[16];
#pragma unroll
  for (int e = 0; e < 16; ++e) {
    int i = lane * 16 + e;
    if (i < 500) {
      float mid = 0.5f * (sVf[i] + sVf[i + 1]);
      kk[e] = (1.f - mid) / (1.f + mid);          // lam = dt/dx = 1 exactly
      vv[e] = sinf((float)(i + 1) * (PI_F / 500.f)); // v0 = fx[1:]
    } else {                                      // spatial padding 500..511
      kk[e] = 0.f; vv[e] = 0.f;
    }
    pp[e] = (e == 0) ? -kk[0] : pp[e - 1] * (-kk[e]);
  }
  float k0 = __int_as_float(__builtin_amdgcn_readfirstlane(__float_as_int(kk[0])));

  // boundary forcing table: tb[t] = gt[t] - gt[t+1]*k0,  gt[t] = sin(t*pi/1000)
  for (int t = lane; t < 500; t += 32) {
    float g0 = sinf((float)t       * (PI_F / 1000.f));
    float g1 = sinf((float)(t + 1) * (PI_F / 1000.f));
    sTb[t] = g0 - g1 * k0;
  }
  // output edges: row X=0 is gt[5T] (gt[0]==fx[0]==0); col T=0 is fx[5X]
  for (int T = lane; T <= 100; T += 32)
    out[ob + T] = sinf((float)T * (PI_F / 200.f));
  for (int X = lane + 1; X <= 100; X += 32)       // lane+1 skips X=0
    out[ob + (size_t)X * 101] = sinf((float)X * (PI_F / 100.f));
  __syncthreads();

  // ---- 500 sequential time steps; spatial recurrence as affine wave scan ----
  for (int t = 0; t < 500; ++t) {
    float vleft = lane_up(vv[15], 1, 0.f, lane);  // v[i-1] across lane boundary
    float tb = sTb[t];
    float prev = vleft;
#pragma unroll
    for (int e = 0; e < 16; ++e) {
      float buf = fmaf(kk[e], vv[e], prev);       // t_D*v (bidiagonal)
      if (lane == 0 && e == 0) buf += tb;         // boundary source
      prev = vv[e];
      ss[e] = (e == 0) ? buf : fmaf(-kk[e], ss[e - 1], buf); // zero-carry scan
    }
    // inter-lane affine scan of (P,S): c_j = S_j + P_j * c_{j-1}
    float S = ss[15], P = pp[15];
#pragma unroll
    for (int d = 1; d < 32; d <<= 1) {
      float Su = lane_up(S, d, 0.f, lane);
      float Pu = lane_up(P, d, 1.f, lane);
      S = fmaf(P, Su, S);
      P = P * Pu;
    }
    float cin = lane_up(S, 1, 0.f, lane);         // exclusive carry-in
#pragma unroll
    for (int e = 0; e < 16; ++e)
      vv[e] = fmaf(pp[e], cin, ss[e]);            // v' = s + p*c_in

    if ((t % 5) == 4) {                           // down-sampled time column
      int T = (t + 1) / 5;
#pragma unroll
      for (int e = 0; e < 16; ++e) {
        int i = lane * 16 + e;
        if (i < 500 && (i % 5) == 4) {            // fine row 5X -> v index 5X-1
          int X = (i + 1) / 5;
          out[ob + (size_t)X * 101 + T] = vv[e];
        }
      }
    }
  }
}

// ---------------------------------------------------------------------------
extern "C" void kernel_launch(void* const* d_in, const int* in_sizes, int n_in,
                              void* d_out, int out_size, void* d_ws, size_t ws_size,
                              hipStream_t stream) {
  (void)in_sizes; (void)n_in; (void)out_size; (void)ws_size;
  const float* vxs = (const float*)d_in[0];       // [64,101] f32
  float* Vf  = (float*)d_ws;                      // [64,512] f32 = 128 KB scratch
  float* out = (float*)d_out;                     // [64,101,101] f32

  cvc_interp_wmma<<<dim3(128), dim3(32), 0, stream>>>(vxs, Vf);
  cvc_solver<<<dim3(64), dim3(32), 0, stream>>>(Vf, out);
}